// GCN_FGGP_33337536151950
// MI455X (gfx1250) — compile-verified
//
#include <hip/hip_runtime.h>
#include <math.h>

typedef __attribute__((ext_vector_type(2))) float v2f;
typedef __attribute__((ext_vector_type(8))) float v8f;
typedef int v4i __attribute__((vector_size(16)));   // matches builtin param type

// ---- CDNA5 async copy-to-LDS availability --------------------------------
#if defined(__has_builtin)
#if __has_builtin(__builtin_amdgcn_global_load_async_to_lds_b128) && \
    __has_builtin(__builtin_amdgcn_s_wait_asynccnt)
#define GCN_ASYNC 1
#endif
#endif
#ifndef GCN_ASYNC
#define GCN_ASYNC 0
#endif

#if GCN_ASYNC
// param0: v4i addrspace(1)* (non-const), param1: v4i addrspace(3)*
#define ASYNC_CP_F4(gsrc, ldst)                                                \
  __builtin_amdgcn_global_load_async_to_lds_b128(                              \
      (__attribute__((address_space(1))) v4i*)(void*)(gsrc),                   \
      (__attribute__((address_space(3))) v4i*)(void*)(ldst), 0, 0)
#endif

// ---------------- degree / normalization ----------------
__global__ void deg_init_k(float* deg, int N) {
  int i = blockIdx.x * blockDim.x + threadIdx.x;
  if (i < N) deg[i] = 1.0f;                    // self-loop contributes 1
}
__global__ void deg_edges_k(const int* __restrict__ dst, float* deg, int E) {
  int e = blockIdx.x * blockDim.x + threadIdx.x;
  if (e < E) atomicAdd(&deg[dst[e]], 1.0f);
}
__global__ void deg_inv_k(float* deg, int N) {
  int i = blockIdx.x * blockDim.x + threadIdx.x;
  if (i < N) deg[i] = rsqrtf(deg[i]);          // deg >= 1 always
}

// ---------------- f32 WMMA GEMM: T[N,128] = X[N,128] @ W[128,128] ----------------
// Block = 256 threads = 8 waves; wave w owns output columns [16w, 16w+16).
// B fragments for the whole K=128 are preloaded into 64 VGPRs per wave; the
// block grid-strides over 16-row A tiles. A tiles are staged in LDS (row
// stride 132 floats -> conflict-free ds reads); with GCN_ASYNC the staging is
// double-buffered via GLOBAL_LOAD_ASYNC_TO_LDS_B128 overlapped with WMMAs.
__global__ void __launch_bounds__(256) gemm128_wmma(const float* __restrict__ X,
                                                    const float* __restrict__ W,
                                                    float* __restrict__ T, int N) {
#if GCN_ASYNC
  __shared__ float lA[2 * 16 * 132];
#else
  __shared__ float lA[16 * 132];
#endif
  const int tid = threadIdx.x;
  const int wave = tid >> 5;
  const int lane = tid & 31;
  const int colTile = wave << 4;
  const int m  = lane & 15;            // fragment row / B,C column
  const int kb = (lane >> 4) << 1;     // 0 for lanes 0-15, 2 for lanes 16-31

  // B fragment kk covers K = 4kk..4kk+3: vgpr0 = row 4kk+kb, vgpr1 = row 4kk+kb+1
  v2f bfrag[32];
#pragma unroll
  for (int kk = 0; kk < 32; ++kk) {
    int k0 = (kk << 2) + kb;
    bfrag[kk].x = W[(k0 + 0) * 128 + colTile + m];
    bfrag[kk].y = W[(k0 + 1) * 128 + colTile + m];
  }

  const int ntiles = (N + 15) >> 4;
  // staging coords: 512 float4s per 16x128 tile, 2 per thread
  const int r0 = tid >> 5,         c0 = (tid & 31) << 2;          // rows 0..7
  const int r1 = (tid + 256) >> 5, c1 = (tid & 31) << 2;          // rows 8..15

#if GCN_ASYNC
  int buf = 0;
  const int rt0 = blockIdx.x;
  if (rt0 < ntiles) {                                  // preload tile rt0 -> buf 0
    int g0 = min(rt0 * 16 + r0, N - 1);
    int g1 = min(rt0 * 16 + r1, N - 1);
    ASYNC_CP_F4(X + g0 * 128 + c0, &lA[r0 * 132 + c0]);
    ASYNC_CP_F4(X + g1 * 128 + c1, &lA[r1 * 132 + c1]);
  }
  for (int rt = rt0; rt < ntiles; rt += gridDim.x) {
    const int nxt = rt + gridDim.x;
    __syncthreads();                                   // all waves done reading buf^1
    if (nxt < ntiles) {                                // prefetch next tile -> buf^1
      const int pbase = (buf ^ 1) * 2112;
      int g0 = min(nxt * 16 + r0, N - 1);
      int g1 = min(nxt * 16 + r1, N - 1);
      ASYNC_CP_F4(X + g0 * 128 + c0, &lA[pbase + r0 * 132 + c0]);
      ASYNC_CP_F4(X + g1 * 128 + c1, &lA[pbase + r1 * 132 + c1]);
      __builtin_amdgcn_s_wait_asynccnt(2);             // oldest 2 (tile rt) complete
    } else {
      __builtin_amdgcn_s_wait_asynccnt(0);
    }
    __syncthreads();                                   // tile rt visible to all waves
    const int abase = buf * 2112;
    v8f acc = {};
#pragma unroll
    for (int kk = 0; kk < 32; ++kk) {
      int k0 = (kk << 2) + kb;
      v2f a;
      a.x = lA[abase + m * 132 + k0];
      a.y = lA[abase + m * 132 + k0 + 1];
      acc = __builtin_amdgcn_wmma_f32_16x16x4_f32(false, a, false, bfrag[kk],
                                                  (short)0, acc, false, false);
    }
    const int row0 = rt << 4;
#pragma unroll
    for (int v = 0; v < 8; ++v) {
      int gr = row0 + v + ((lane >> 4) << 3);
      if (gr < N) T[gr * 128 + colTile + m] = acc[v];
    }
    buf ^= 1;
  }
#else
  for (int rt = blockIdx.x; rt < ntiles; rt += gridDim.x) {
    const int row0 = rt << 4;
    __syncthreads();
    {
      int g0 = min(row0 + r0, N - 1);
      int g1 = min(row0 + r1, N - 1);
      *(float4*)&lA[r0 * 132 + c0] = *(const float4*)&X[g0 * 128 + c0];
      *(float4*)&lA[r1 * 132 + c1] = *(const float4*)&X[g1 * 128 + c1];
    }
    __syncthreads();
    v8f acc = {};
#pragma unroll
    for (int kk = 0; kk < 32; ++kk) {
      int k0 = (kk << 2) + kb;
      v2f a;
      a.x = lA[m * 132 + k0];
      a.y = lA[m * 132 + k0 + 1];
      acc = __builtin_amdgcn_wmma_f32_16x16x4_f32(false, a, false, bfrag[kk],
                                                  (short)0, acc, false, false);
    }
#pragma unroll
    for (int v = 0; v < 8; ++v) {
      int gr = row0 + v + ((lane >> 4) << 3);
      if (gr < N) T[gr * 128 + colTile + m] = acc[v];
    }
  }
#endif
}

// ---------------- aggregation ----------------
// A[i,:] = T[i,:]*dinv[i]^2 + bias   (self-loop message + bias); 4 feats/thread
__global__ void agg_self_k(const float* __restrict__ T, const float* __restrict__ dinv,
                           const float* __restrict__ bias, float* __restrict__ A,
                           int nquads) {                      // nquads = N*32
  int idx = blockIdx.x * blockDim.x + threadIdx.x;
  if (idx >= nquads) return;
  int i = idx >> 5, f4 = (idx & 31) << 2;
  float d = dinv[i];
  float w = d * d;
  float4 t = *(const float4*)&T[i * 128 + f4];
  float4 b = *(const float4*)&bias[f4];
  float4 o = {t.x * w + b.x, t.y * w + b.y, t.z * w + b.z, t.w * w + b.w};
  *(float4*)&A[i * 128 + f4] = o;
}
// A[dst,:] += T[src,:] * dinv[src]*dinv[dst]; 4 feats/thread, B128 gather
__global__ void agg_edges_k(const float* __restrict__ T, const int* __restrict__ src,
                            const int* __restrict__ dst, const float* __restrict__ dinv,
                            float* __restrict__ A, int nquads) {   // nquads = E*32
  int idx = blockIdx.x * blockDim.x + threadIdx.x;
  if (idx >= nquads) return;
  int e = idx >> 5, f4 = (idx & 31) << 2;
  int s = src[e], d = dst[e];
  float w = dinv[s] * dinv[d];
  float4 t = *(const float4*)&T[s * 128 + f4];
  float* out = &A[d * 128 + f4];
  atomicAdd(out + 0, t.x * w);
  atomicAdd(out + 1, t.y * w);
  atomicAdd(out + 2, t.z * w);
  atomicAdd(out + 3, t.w * w);
}
__global__ void relu_k(float* __restrict__ A, int nquads) {
  int idx = blockIdx.x * blockDim.x + threadIdx.x;
  if (idx >= nquads) return;
  float4* p = (float4*)A + idx;
  float4 v = *p;
  v.x = fmaxf(v.x, 0.f); v.y = fmaxf(v.y, 0.f);
  v.z = fmaxf(v.z, 0.f); v.w = fmaxf(v.w, 0.f);
  *p = v;
}

// ---------------- classifier + log_softmax / softmax ----------------
// 320 threads = 8 nodes x 40 outputs per block; feature rows staged in LDS.
__global__ void __launch_bounds__(320) classifier_k(const float* __restrict__ feat,
                                                    const float* __restrict__ Wc,
                                                    const float* __restrict__ bc,
                                                    float* __restrict__ out_ls,
                                                    float* __restrict__ out_sm, int N) {
  __shared__ float sFeat[8][128];
  __shared__ float sLog[8][40];
  const int tid = threadIdx.x;
  const int ln = tid / 40;              // local node 0..7
  const int j  = tid % 40;              // output class
  const int node0 = blockIdx.x * 8;

  for (int idx = tid; idx < 8 * 128; idx += 320) {
    int r = idx >> 7, c = idx & 127;
    int g = node0 + r;
    sFeat[r][c] = (g < N) ? feat[g * 128 + c] : 0.0f;
  }
  __syncthreads();

  float acc = bc[j];
#pragma unroll 8
  for (int k = 0; k < 128; ++k) acc += sFeat[ln][k] * Wc[k * 40 + j];
  sLog[ln][j] = acc;
  __syncthreads();

  int node = node0 + ln;
  if (node < N) {
    float mx = -INFINITY;
    for (int t = 0; t < 40; ++t) mx = fmaxf(mx, sLog[ln][t]);
    float sum = 0.0f;
    for (int t = 0; t < 40; ++t) sum += expf(sLog[ln][t] - mx);
    float lse = mx + logf(sum);
    out_ls[node * 40 + j] = acc - lse;
    out_sm[node * 40 + j] = expf(acc - lse);
  }
}

// ---------------- launch ----------------
extern "C" void kernel_launch(void* const* d_in, const int* in_sizes, int n_in,
                              void* d_out, int out_size, void* d_ws, size_t ws_size,
                              hipStream_t stream) {
  const float* x  = (const float*)d_in[0];
  const int*   ei = (const int*)d_in[1];
  const float* W1 = (const float*)d_in[2];
  const float* b1 = (const float*)d_in[3];
  const float* W2 = (const float*)d_in[4];
  const float* b2 = (const float*)d_in[5];
  const float* Wc = (const float*)d_in[6];
  const float* bc = (const float*)d_in[7];

  const int N = in_sizes[0] / 128;
  const int E = in_sizes[1] / 2;
  const int* src = ei;
  const int* dst = ei + E;

  // d_out = [log_softmax N*40 | features N*128 | softmax N*40]
  float* out_ls = (float*)d_out;
  float* feat   = out_ls + (size_t)N * 40;
  float* conf   = feat + (size_t)N * 128;

  float* dinv = (float*)d_ws;
  float* T    = dinv + N;                 // N*128 scratch (x@W, h@W)
  float* H    = T + (size_t)N * 128;      // N*128 hidden activations

  const int NQ = N * 32;                  // node-feature float4s (3.2M)
  const int EQ = E * 32;                  // edge-feature float4s (51.2M)
  const int TB = 256;
  const int gN  = (N + TB - 1) / TB;
  const int gE  = (E + TB - 1) / TB;
  const int gNQ = (NQ + TB - 1) / TB;
  const int gEQ = (EQ + TB - 1) / TB;

  // degrees -> dinv
  deg_init_k<<<gN, TB, 0, stream>>>(dinv, N);
  deg_edges_k<<<gE, TB, 0, stream>>>(dst, dinv, E);
  deg_inv_k<<<gN, TB, 0, stream>>>(dinv, N);

  // layer 1: T = x@W1 ; H = aggregate(T) + b1 ; relu
  gemm128_wmma<<<1024, 256, 0, stream>>>(x, W1, T, N);
  agg_self_k<<<gNQ, TB, 0, stream>>>(T, dinv, b1, H, NQ);
  agg_edges_k<<<gEQ, TB, 0, stream>>>(T, src, dst, dinv, H, EQ);
  relu_k<<<gNQ, TB, 0, stream>>>(H, NQ);

  // layer 2: T = H@W2 ; features = aggregate(T) + b2 (directly into d_out)
  gemm128_wmma<<<1024, 256, 0, stream>>>(H, W2, T, N);
  agg_self_k<<<gNQ, TB, 0, stream>>>(T, dinv, b2, feat, NQ);
  agg_edges_k<<<gEQ, TB, 0, stream>>>(T, src, dst, dinv, feat, EQ);

  // classifier + softmax/log_softmax
  classifier_k<<<(N + 7) / 8, 320, 0, stream>>>(feat, Wc, bc, out_ls, conf, N);
}